// RetIQANet_27977416966591
// MI455X (gfx1250) — compile-verified
//
#include <hip/hip_runtime.h>
#include <hip/hip_bf16.h>

typedef __attribute__((ext_vector_type(2))) float v2f;
typedef __attribute__((ext_vector_type(8))) float v8f;

#define B_   64
#define FC_  8192
#define FD_  2048
#define N_   1000
#define D_   25
#define K_   9

// ---------------------------------------------------------------------------
// Kernel 1: inv_sem[n] = rsqrt(sum(sem[n,:]^2))   (1000 rows x 8192)
// ---------------------------------------------------------------------------
__global__ __launch_bounds__(256) void sem_norms_kernel(
    const float* __restrict__ sem, float* __restrict__ inv_sem)
{
    __shared__ float sh[256];
    const int n = blockIdx.x;
    const int t = threadIdx.x;
    const float* row = sem + (size_t)n * FC_;
    float s = 0.f;
    for (int i = t; i < FC_; i += 256) { float v = row[i]; s += v * v; }
    sh[t] = s;
    __syncthreads();
    for (int off = 128; off > 0; off >>= 1) {
        if (t < off) sh[t] += sh[t + off];
        __syncthreads();
    }
    if (t == 0) inv_sem[n] = rsqrtf(sh[0]);
}

// ---------------------------------------------------------------------------
// Kernel 2: raw dots sem_dot[b,n] = fc[b,:] . sem[n,:]  via V_WMMA_F32_16X16X4_F32
// One wave per 16x16 output tile. grid = (63 ntiles, 4 mtiles).
// A (16x4 f32): lanes 0-15 hold M=lane, K=0..1; lanes 16-31 hold M=lane-16, K=2..3.
// B (4x16 f32): lanes 0-15 hold N=lane, K=0..1; lanes 16-31 hold N=lane-16, K=2..3.
// C/D (16x16 f32): VGPR r -> row M = r + 8*(lane>=16), col N = lane&15.
//
// N=1000 tail handling: column n of D depends only on column n of B, so lanes
// with bcol >= N just clamp their row pointer to a valid row (no OOB) and the
// garbage columns are discarded by the store-side guard. This keeps the inner
// loop branch-free: 2x global_load_b64 + 1x v_wmma per K=4 step.
// ---------------------------------------------------------------------------
__global__ __launch_bounds__(32) void gemm_sem_kernel(
    const float* __restrict__ fc, const float* __restrict__ sem,
    float* __restrict__ sem_dot)
{
    const int lane  = threadIdx.x;      // 0..31 (wave32)
    const int ntile = blockIdx.x;       // 0..62
    const int mtile = blockIdx.y;       // 0..3
    const int m0 = mtile * 16;
    const int n0 = ntile * 16;
    const int half = lane >> 4;         // 0: K 0-1, 1: K 2-3
    const int lid  = lane & 15;

    const int arow   = m0 + lid;
    const int bcol   = n0 + lid;
    const bool bvalid = (bcol < N_);
    const int bsafe  = bvalid ? bcol : (N_ - 1);   // clamp: no OOB, no branch in loop

    // 8-byte aligned float2 streams (half*2 floats offset, k steps by 4)
    const float2* a2 = (const float2*)(fc  + (size_t)arow  * FC_ + half * 2);
    const float2* b2 = (const float2*)(sem + (size_t)bsafe * FC_ + half * 2);

    v8f acc0 = {};
    v8f acc1 = {};

    #pragma unroll 4
    for (int k = 0; k < FC_; k += 8) {
        // step k
        float2 af0 = a2[k >> 1];
        float2 bf0 = b2[k >> 1];
        v2f a0; a0[0] = af0.x; a0[1] = af0.y;
        v2f b0; b0[0] = bf0.x; b0[1] = bf0.y;
        acc0 = __builtin_amdgcn_wmma_f32_16x16x4_f32(
            false, a0, false, b0, (short)0, acc0, false, false);
        // step k+4 (independent accumulator for ILP across the WMMA pipe)
        float2 af1 = a2[(k + 4) >> 1];
        float2 bf1 = b2[(k + 4) >> 1];
        v2f a1; a1[0] = af1.x; a1[1] = af1.y;
        v2f b1; b1[0] = bf1.x; b1[1] = bf1.y;
        acc1 = __builtin_amdgcn_wmma_f32_16x16x4_f32(
            false, a1, false, b1, (short)0, acc1, false, false);
    }

    if (bvalid) {
        #pragma unroll
        for (int r = 0; r < 8; ++r) {
            const int m = m0 + r + half * 8;
            sem_dot[(size_t)m * N_ + bcol] = acc0[r] + acc1[r];
        }
    }
}

// ---------------------------------------------------------------------------
// Kernel 3: top-9 per row of score[b,n] = sem_dot[b,n] * inv_sem[n]
// (|fc[b]| cancels). JAX top_k tie-break: lower index wins on equal value.
// ---------------------------------------------------------------------------
__global__ __launch_bounds__(256) void topk9_kernel(
    const float* __restrict__ sem_dot, const float* __restrict__ inv_sem,
    int* __restrict__ top_idx)
{
    __shared__ float s[N_];
    __shared__ float rv[256];
    __shared__ int   ri[256];
    const int b = blockIdx.x;
    const int t = threadIdx.x;

    for (int n = t; n < N_; n += 256)
        s[n] = sem_dot[(size_t)b * N_ + n] * inv_sem[n];
    __syncthreads();

    for (int k = 0; k < K_; ++k) {
        float bv = -__builtin_inff();
        int   bi = N_;
        for (int n = t; n < N_; n += 256) {
            float v = s[n];
            if (v > bv || (v == bv && n < bi)) { bv = v; bi = n; }
        }
        rv[t] = bv; ri[t] = bi;
        __syncthreads();
        for (int off = 128; off > 0; off >>= 1) {
            if (t < off) {
                float ov = rv[t + off]; int oi = ri[t + off];
                if (ov > rv[t] || (ov == rv[t] && oi < ri[t])) {
                    rv[t] = ov; ri[t] = oi;
                }
            }
            __syncthreads();
        }
        if (t == 0) {
            top_idx[b * K_ + k] = ri[0];
            s[ri[0]] = -__builtin_inff();   // remove winner
        }
        __syncthreads();
    }
}

// ---------------------------------------------------------------------------
// Kernel 4: dscore[b,k,d] = (fd[b] . v) * rsqrt(|v|^2),  v = dst_fs[idx,d]
// (|fd[b]| cancels in the argmax over d). One wave per (b,k,d).
// ---------------------------------------------------------------------------
__global__ __launch_bounds__(32) void dscore_kernel(
    const float* __restrict__ fd, const float* __restrict__ dst_fs,
    const int* __restrict__ top_idx, float* __restrict__ dscore)
{
    const int d    = blockIdx.x;   // 0..24
    const int k    = blockIdx.y;   // 0..8
    const int b    = blockIdx.z;   // 0..63
    const int lane = threadIdx.x;

    const int ref = top_idx[b * K_ + k];
    const float* v = dst_fs + ((size_t)ref * D_ + d) * FD_;
    const float* q = fd + (size_t)b * FD_;

    float dot = 0.f, n2 = 0.f;
    for (int i = lane; i < FD_; i += 32) {
        float x = v[i];
        dot += q[i] * x;
        n2  += x * x;
    }
    #pragma unroll
    for (int off = 16; off > 0; off >>= 1) {
        dot += __shfl_xor(dot, off, 32);
        n2  += __shfl_xor(n2,  off, 32);
    }
    if (lane == 0)
        dscore[((size_t)b * K_ + k) * D_ + d] = dot * rsqrtf(n2);
}

// ---------------------------------------------------------------------------
// Kernel 5: per (b,k) argmax over d (first-max, like jnp.argmax), gather
// metric, mean over k. One wave per b.
// ---------------------------------------------------------------------------
__global__ __launch_bounds__(32) void finalize_kernel(
    const float* __restrict__ dscore, const int* __restrict__ top_idx,
    const float* __restrict__ metrics, float* __restrict__ out)
{
    const int b    = blockIdx.x;
    const int lane = threadIdx.x;
    float val = 0.f;
    if (lane < K_) {
        const float* sc = dscore + ((size_t)b * K_ + lane) * D_;
        float bv = sc[0];
        int   bd = 0;
        for (int d = 1; d < D_; ++d) {
            float v = sc[d];
            if (v > bv) { bv = v; bd = d; }
        }
        const int ref = top_idx[b * K_ + lane];
        val = metrics[(size_t)ref * D_ + bd];
    }
    #pragma unroll
    for (int off = 16; off > 0; off >>= 1) val += __shfl_xor(val, off, 32);
    if (lane == 0) out[b] = val * (1.f / 9.f);
}

// ---------------------------------------------------------------------------
extern "C" void kernel_launch(void* const* d_in, const int* in_sizes, int n_in,
                              void* d_out, int out_size, void* d_ws, size_t ws_size,
                              hipStream_t stream)
{
    const float* fc   = (const float*)d_in[0];   // [64, 8192]
    const float* fd   = (const float*)d_in[1];   // [64, 2048]
    const float* sem  = (const float*)d_in[2];   // [1000, 8192]
    const float* dst  = (const float*)d_in[3];   // [1000, 25, 2048]
    const float* met  = (const float*)d_in[4];   // [1000, 25]
    float* out = (float*)d_out;                  // [64]

    // workspace carving (~320 KB)
    float* inv_sem = (float*)d_ws;                       // 1024 floats (1000 used)
    float* sem_dot = inv_sem + 1024;                     // 64*1000 floats
    int*   top_idx = (int*)(sem_dot + B_ * N_);          // 64*9 ints
    float* dscore  = (float*)(top_idx + B_ * K_);        // 64*9*25 floats

    sem_norms_kernel<<<N_, 256, 0, stream>>>(sem, inv_sem);
    gemm_sem_kernel<<<dim3((N_ + 15) / 16, B_ / 16), 32, 0, stream>>>(fc, sem, sem_dot);
    topk9_kernel<<<B_, 256, 0, stream>>>(sem_dot, inv_sem, top_idx);
    dscore_kernel<<<dim3(D_, K_, B_), 32, 0, stream>>>(fd, dst, top_idx, dscore);
    finalize_kernel<<<B_, 32, 0, stream>>>(dscore, top_idx, met, out);
}